// LiftSplatShoot_76089640616461
// MI455X (gfx1250) — compile-verified
//
#include <hip/hip_runtime.h>
#include <hip/hip_bf16.h>
#include <math.h>

#define Bn 12          // B*N
#define Cc 3
#define Hh 64
#define Ww 176
#define Dd 64
#define HW (Hh*Ww)     // 11264

typedef float v2f __attribute__((ext_vector_type(2)));
typedef float v8f __attribute__((ext_vector_type(8)));

// ---------------- Kernel 1: softmax row stats (max, sum of exp) ----------------
// One block per (bn, c) row of length HW. 36 blocks x 256 threads.
__global__ void lss_row_stats(const float* __restrict__ feat, float* __restrict__ stats) {
    const int row = blockIdx.x;                 // 0..35  == bn*3 + c
    const float* f = feat + (size_t)row * HW;
    __shared__ float red[256];
    const int tid = threadIdx.x;

    float mx = -INFINITY;
    for (int p = tid; p < HW; p += 256) mx = fmaxf(mx, f[p]);
    red[tid] = mx; __syncthreads();
    for (int s = 128; s > 0; s >>= 1) {
        if (tid < s) red[tid] = fmaxf(red[tid], red[tid + s]);
        __syncthreads();
    }
    mx = red[0]; __syncthreads();

    float sm = 0.0f;
    for (int p = tid; p < HW; p += 256) sm += expf(f[p] - mx);
    red[tid] = sm; __syncthreads();
    for (int s = 128; s > 0; s >>= 1) {
        if (tid < s) red[tid] += red[tid + s];
        __syncthreads();
    }
    if (tid == 0) { stats[row * 2] = mx; stats[row * 2 + 1] = red[0]; }
}

// ---------------- Kernel 2: per-pixel scalar s[bn,i,p] ----------------
// s_i = sum_j E[i,j] * (K^-1 [x y 1])_j * prob_j      (1.6 MB workspace)
// Grid: (HW/256 = 44, Bn = 12), 256 threads.
__global__ void lss_pixel(const float* __restrict__ feat,
                          const float* __restrict__ intr,
                          const float* __restrict__ extr,
                          const float* __restrict__ xy1,
                          const float* __restrict__ stats,
                          float* __restrict__ sarr) {
    const int bn = blockIdx.y;
    const int p  = blockIdx.x * 256 + threadIdx.x;   // 44*256 == HW exactly
    const float x = xy1[p];
    const float y = xy1[HW + p];

    // 3x3 inverse of upper-left intrinsics block (adjugate / det)
    const float* M = intr + bn * 16;
    const float m00 = M[0], m01 = M[1], m02 = M[2];
    const float m10 = M[4], m11 = M[5], m12 = M[6];
    const float m20 = M[8], m21 = M[9], m22 = M[10];
    const float c00 = m11 * m22 - m12 * m21;
    const float c01 = m12 * m20 - m10 * m22;
    const float c02 = m10 * m21 - m11 * m20;
    const float id  = 1.0f / (m00 * c00 + m01 * c01 + m02 * c02);
    const float i00 = c00 * id;
    const float i01 = (m02 * m21 - m01 * m22) * id;
    const float i02 = (m01 * m12 - m02 * m11) * id;
    const float i10 = c01 * id;
    const float i11 = (m00 * m22 - m02 * m20) * id;
    const float i12 = (m02 * m10 - m00 * m12) * id;
    const float i20 = c02 * id;
    const float i21 = (m01 * m20 - m00 * m21) * id;
    const float i22 = (m00 * m11 - m01 * m10) * id;

    float r0 = i00 * x + i01 * y + i02;
    float r1 = i10 * x + i11 * y + i12;
    float r2 = i20 * x + i21 * y + i22;

    float t[3];
    const float rr[3] = { r0, r1, r2 };
#pragma unroll
    for (int j = 0; j < 3; ++j) {
        const int row = bn * 3 + j;
        const float f = feat[(size_t)row * HW + p];
        const float prob = expf(f - stats[row * 2]) / stats[row * 2 + 1];
        t[j] = rr[j] * prob;
    }

    const float* E = extr + bn * 16;
#pragma unroll
    for (int i = 0; i < 3; ++i) {
        const float si = E[i * 4 + 0] * t[0] + E[i * 4 + 1] * t[1] + E[i * 4 + 2] * t[2];
        sarr[(size_t)(bn * 3 + i) * HW + p] = si;
    }
}

// ---------------- Kernel 3: WMMA outer-product expansion (bandwidth kernel) ----
// out[R, d] = sarr[R] * depths[d] + bias(R),  R = (bn*3+i)*HW + p, 64 d's per row.
// One wave -> 16 consecutive rows x 64 depths via 4x V_WMMA_F32_16X16X4_F32.
// A(16x4): col0 = s, rest 0.  B(4x16): row0 = depths chunk, rest 0.  C = bias.
// 3168 blocks x 256 threads (8 waves) covers 405,504 rows exactly; EXEC all-1s.
__global__ void lss_expand_wmma(const float* __restrict__ sarr,
                                const float* __restrict__ extr,
                                const float* __restrict__ depths,
                                float* __restrict__ out) {
    const int wave = threadIdx.x >> 5;
    const int lane = threadIdx.x & 31;
    const int hi   = lane >> 4;        // 0: lanes 0-15 (K=0,1), 1: lanes 16-31 (K=2,3)
    const int ln   = lane & 15;
    const int R0   = (blockIdx.x * 8 + wave) * 16;

    // 16-row group never crosses an (bn,i) boundary (HW % 16 == 0) -> uniform bias
    const int bni = R0 / HW;           // 0..35
    const int bn  = bni / 3;
    const int ci  = bni - bn * 3;
    const float bias = extr[bn * 16 + ci * 4 + 3];

    const float sval = sarr[R0 + ln];
    v2f a; a.x = hi ? 0.0f : sval; a.y = 0.0f;   // A[m][0] = s, other K zero

    v8f c;
#pragma unroll
    for (int v = 0; v < 8; ++v) c[v] = bias;

    float* base = out + (size_t)R0 * Dd;

#pragma unroll
    for (int t = 0; t < 4; ++t) {
        const int d0 = t * 16;
        const float dep = depths[d0 + ln];
        v2f b; b.x = hi ? 0.0f : dep; b.y = 0.0f; // B[0][n] = depth, other K zero

        // D = A x B + C  (fp32, exact: only K=0 contributes)
        v8f d = __builtin_amdgcn_wmma_f32_16x16x4_f32(
            /*neg_a=*/false, a, /*neg_b=*/false, b,
            /*c_mod=*/(short)0, c, /*reuse_a=*/false, /*reuse_b=*/false);

        // D layout: VGPR v -> row M = v + 8*hi, col N = ln
#pragma unroll
        for (int v = 0; v < 8; ++v) {
            base[(size_t)(v + hi * 8) * Dd + d0 + ln] = d[v];
        }
    }
}

extern "C" void kernel_launch(void* const* d_in, const int* in_sizes, int n_in,
                              void* d_out, int out_size, void* d_ws, size_t ws_size,
                              hipStream_t stream) {
    (void)in_sizes; (void)n_in; (void)out_size; (void)ws_size;
    const float* feat   = (const float*)d_in[0];  // (B*N, C, H, W)
    const float* intr   = (const float*)d_in[1];  // (B, N, 4, 4)
    const float* extr   = (const float*)d_in[2];  // (B, N, 4, 4)
    const float* xy1    = (const float*)d_in[3];  // (3, HW)
    const float* depths = (const float*)d_in[4];  // (D,)
    float* out = (float*)d_out;

    float* stats = (float*)d_ws;            // 72 floats
    float* sarr  = (float*)d_ws + 256;      // 405,504 floats (~1.6 MB)

    lss_row_stats<<<Bn * Cc, 256, 0, stream>>>(feat, stats);

    dim3 g2(HW / 256, Bn);
    lss_pixel<<<g2, 256, 0, stream>>>(feat, intr, extr, xy1, stats, sarr);

    const int rows = Bn * Cc * HW;          // 405,504
    lss_expand_wmma<<<rows / 128, 256, 0, stream>>>(sarr, extr, depths, out);
}